// AdvancedWeightPredictorNetwork_317827580067
// MI455X (gfx1250) — compile-verified
//
#include <hip/hip_runtime.h>
#include <math.h>

#define BDIM 8192
#define FDIM 256
#define CDIM 8
#define KNN  5
#define HDIM 64
#define ODIM 32

typedef float v2f __attribute__((ext_vector_type(2)));
typedef float v4f __attribute__((ext_vector_type(4)));
typedef float v8f __attribute__((ext_vector_type(8)));

// ---------------------------------------------------------------------------
// Kernel 1: center norms + inter-cluster distance sum (tiny, 1 block)
// ---------------------------------------------------------------------------
__global__ void k_centers(const float* __restrict__ cc,
                          float* __restrict__ cnorm,
                          float* __restrict__ inter_sum) {
  __shared__ float red[64];
  const int t = threadIdx.x;   // 64 threads
  if (t < CDIM) {
    float s = 0.f;
    for (int f = 0; f < FDIM; ++f) { float v = cc[t * FDIM + f]; s += v * v; }
    cnorm[t] = s;
  }
  const int i = t / CDIM, j = t % CDIM;
  float d = 0.f;
  if (i != j) {
    float s = 0.f;
    for (int f = 0; f < FDIM; ++f) {
      float dv = cc[i * FDIM + f] - cc[j * FDIM + f];
      s += dv * dv;
    }
    d = sqrtf(fmaxf(s, 1e-12f));
  }
  red[t] = d;
  __syncthreads();
  for (int off = 32; off > 0; off >>= 1) {
    if (t < off) red[t] += red[t + off];
    __syncthreads();
  }
  if (t == 0) inter_sum[0] = red[0];
}

// ---------------------------------------------------------------------------
// Kernel 2: per-row mean / std(ddof=1) / entropy(softmax) / squared norm
// one wave (32 lanes) per row, 8 rows per 256-thread block
// ---------------------------------------------------------------------------
__global__ void k_rowstats(const float* __restrict__ x,
                           float* __restrict__ stats_out,
                           float* __restrict__ xnorm2) {
  const int row  = blockIdx.x * (blockDim.x >> 5) + (threadIdx.x >> 5);
  const int lane = threadIdx.x & 31;
  if (row >= BDIM) return;
  const float* xr = x + (size_t)row * FDIM;

  float xs[8];
  float s1 = 0.f, s2 = 0.f, mx = -__builtin_inff();
#pragma unroll
  for (int k = 0; k < 8; ++k) {
    float v = xr[lane + 32 * k];
    xs[k] = v; s1 += v; s2 += v * v; mx = fmaxf(mx, v);
  }
  for (int off = 16; off > 0; off >>= 1) {
    s1 += __shfl_xor(s1, off, 32);
    s2 += __shfl_xor(s2, off, 32);
    mx  = fmaxf(mx, __shfl_xor(mx, off, 32));
  }
  float Z = 0.f, Sx = 0.f;
#pragma unroll
  for (int k = 0; k < 8; ++k) {
    float tshift = xs[k] - mx;
    float e = expf(tshift);
    Z += e; Sx += tshift * e;
  }
  for (int off = 16; off > 0; off >>= 1) {
    Z  += __shfl_xor(Z, off, 32);
    Sx += __shfl_xor(Sx, off, 32);
  }
  if (lane == 0) {
    float mean = s1 / (float)FDIM;
    float var  = (s2 - s1 * s1 / (float)FDIM) / (float)(FDIM - 1);
    float stdv = sqrtf(fmaxf(var, 0.f)) + 1e-8f;
    float ent  = logf(Z) - Sx / Z;   // -sum p*logp with max-shift
    stats_out[row * 3 + 0] = mean;
    stats_out[row * 3 + 1] = stdv;
    stats_out[row * 3 + 2] = ent;
    xnorm2[row] = s2;
  }
}

// ---------------------------------------------------------------------------
// Kernel 3: soft assignment + per-row intra-loss partial
// one wave per row; centers cached in LDS
// ---------------------------------------------------------------------------
__global__ void k_assign(const float* __restrict__ x,
                         const float* __restrict__ cc,
                         const float* __restrict__ temp,
                         const float* __restrict__ cw,
                         const float* __restrict__ xnorm2,
                         const float* __restrict__ cnorm,
                         float* __restrict__ assign_out,
                         float* __restrict__ intra_part) {
  __shared__ float shc[CDIM * FDIM];
  __shared__ float shn[CDIM], shw[CDIM];
  const int tid = threadIdx.x;
  for (int i = tid; i < CDIM * FDIM; i += blockDim.x) shc[i] = cc[i];
  if (tid < CDIM) { shn[tid] = cnorm[tid]; shw[tid] = cw[tid]; }
  __syncthreads();

  const int row  = blockIdx.x * (blockDim.x >> 5) + (tid >> 5);
  const int lane = tid & 31;
  if (row >= BDIM) return;
  const float* xr = x + (size_t)row * FDIM;

  float acc[CDIM];
#pragma unroll
  for (int c = 0; c < CDIM; ++c) acc[c] = 0.f;
  for (int f = lane; f < FDIM; f += 32) {
    float xv = xr[f];
#pragma unroll
    for (int c = 0; c < CDIM; ++c) acc[c] += xv * shc[c * FDIM + f];
  }
#pragma unroll
  for (int c = 0; c < CDIM; ++c)
    for (int off = 16; off > 0; off >>= 1) acc[c] += __shfl_xor(acc[c], off, 32);

  if (lane == 0) {
    const float T  = temp[0];
    const float n2 = xnorm2[row];
    float dc[CDIM], lg[CDIM], mxl = -__builtin_inff();
#pragma unroll
    for (int c = 0; c < CDIM; ++c) {
      float d2 = n2 + shn[c] - 2.f * acc[c];
      dc[c] = sqrtf(fmaxf(d2, 1e-12f));
      lg[c] = -dc[c] / T;
      mxl = fmaxf(mxl, lg[c]);
    }
    float Z = 0.f, a[CDIM];
#pragma unroll
    for (int c = 0; c < CDIM; ++c) { a[c] = expf(lg[c] - mxl); Z += a[c]; }
    float ip = 0.f;
#pragma unroll
    for (int c = 0; c < CDIM; ++c) {
      float as = a[c] / Z * shw[c];
      assign_out[row * CDIM + c] = as;
      ip += dc[c] * as;
    }
    intra_part[row] = ip;
  }
}

// ---------------------------------------------------------------------------
// Kernel 4: fused Gram (WMMA f32 16x16x4) + noisy top-5 per row.
// Block = 64 threads (2 waves). Each wave register-blocks 32 rows x 64 cols:
// per k-step 2 A-frags + 4 B-frags (6 LDS loads) feed 8 WMMAs.
// A (64 x 256) stays LDS-resident for the whole block; B chunks (64 x 32)
// are double-buffered, one barrier per chunk. Padded pitches make all
// fragment loads LDS-bank-conflict-free.
// ---------------------------------------------------------------------------
#define IB 64
#define JBW 64
#define KC 32
#define APITCH 260   // 256 + 4: stride-4 bank pattern, 16B-aligned rows
#define BPITCH 36    // 32 + 4:  stride-36 bank pattern, 16B-aligned rows
#define SPITCH 68

struct ShmG {
  float A[IB * APITCH];                 // 66560 B
  union {
    float Bb[2][JBW * BPITCH];          // 18432 B
    float S[IB * SPITCH];               // 17408 B
  } u;
};

__global__ __launch_bounds__(64, 1) void k_gram_topk(
    const float* __restrict__ x, const float* __restrict__ noise,
    const float* __restrict__ xnorm2, float* __restrict__ knn_out) {
  __shared__ ShmG sh;

  const int tid   = threadIdx.x;       // 0..63
  const int lane  = tid & 31;
  const int wv    = tid >> 5;          // wave 0/1 -> 32-row strip
  const int hl    = lane >> 4;         // lane half selects K pair (ISA layout)
  const int ml    = lane & 15;         // M / N index within 16
  const int ibase = blockIdx.x * IB;

  // ---- load the block's A panel (64 x 256) into LDS once ----
#pragma unroll
  for (int i = 0; i < 64; ++i) {
    int idx = tid + 64 * i;            // 0..4095 float4s
    int r   = idx >> 6;                // 64 float4 per row
    int c4  = (idx & 63) * 4;
    *(float4*)&sh.A[r * APITCH + c4] =
        *(const float4*)&x[(size_t)(ibase + r) * FDIM + c4];
  }

  float key[KNN], val[KNN];
#pragma unroll
  for (int i = 0; i < KNN; ++i) { key[i] = __builtin_inff(); val[i] = 0.f; }

  const int   myrow = ibase + tid;
  const float ni    = xnorm2[myrow];
  const float* nzrow = noise + (size_t)myrow * BDIM;

  const int rLo = wv * 32 + ml;        // A rows for this lane (low 16-group)
  const int rHi = rLo + 16;            // high 16-group

  for (int jb = 0; jb < BDIM; jb += JBW) {
    __syncthreads();                   // S-scan / A-panel stores complete
    // prologue: B chunk 0 -> buffer 0
#pragma unroll
    for (int i = 0; i < 8; ++i) {
      int idx = tid + 64 * i;          // 0..511 float4s
      int r   = idx >> 3;              // 8 float4 per row
      int c4  = (idx & 7) * 4;
      *(float4*)&sh.u.Bb[0][r * BPITCH + c4] =
          *(const float4*)&x[(size_t)(jb + r) * FDIM + c4];
    }
    __syncthreads();

    v8f aL0 = {}, aL1 = {}, aL2 = {}, aL3 = {};
    v8f aH0 = {}, aH1 = {}, aH2 = {}, aH3 = {};

    for (int kci = 0; kci < FDIM / KC; ++kci) {
      const int cur = kci & 1;
      if (kci + 1 < FDIM / KC) {       // prefetch next chunk into other buffer
        const int kc = (kci + 1) * KC;
#pragma unroll
        for (int i = 0; i < 8; ++i) {
          int idx = tid + 64 * i;
          int r   = idx >> 3;
          int c4  = (idx & 7) * 4;
          *(float4*)&sh.u.Bb[cur ^ 1][r * BPITCH + c4] =
              *(const float4*)&x[(size_t)(jb + r) * FDIM + kc + c4];
        }
      }
      const float* Bc = sh.u.Bb[cur];
      const int kb = kci * KC;
#pragma unroll
      for (int ks = 0; ks < KC; ks += 4) {
        const int kk = ks + 2 * hl;    // {K,K+1} pair per lane-half
        v2f aLo = *(const v2f*)&sh.A[rLo * APITCH + kb + kk];
        v2f aHi = *(const v2f*)&sh.A[rHi * APITCH + kb + kk];
        v2f b0  = *(const v2f*)&Bc[(0 * 16 + ml) * BPITCH + kk];
        v2f b1  = *(const v2f*)&Bc[(1 * 16 + ml) * BPITCH + kk];
        v2f b2  = *(const v2f*)&Bc[(2 * 16 + ml) * BPITCH + kk];
        v2f b3  = *(const v2f*)&Bc[(3 * 16 + ml) * BPITCH + kk];
        aL0 = __builtin_amdgcn_wmma_f32_16x16x4_f32(false, aLo, false, b0, (short)0, aL0, false, false);
        aL1 = __builtin_amdgcn_wmma_f32_16x16x4_f32(false, aLo, false, b1, (short)0, aL1, false, false);
        aL2 = __builtin_amdgcn_wmma_f32_16x16x4_f32(false, aLo, false, b2, (short)0, aL2, false, false);
        aL3 = __builtin_amdgcn_wmma_f32_16x16x4_f32(false, aLo, false, b3, (short)0, aL3, false, false);
        aH0 = __builtin_amdgcn_wmma_f32_16x16x4_f32(false, aHi, false, b0, (short)0, aH0, false, false);
        aH1 = __builtin_amdgcn_wmma_f32_16x16x4_f32(false, aHi, false, b1, (short)0, aH1, false, false);
        aH2 = __builtin_amdgcn_wmma_f32_16x16x4_f32(false, aHi, false, b2, (short)0, aH2, false, false);
        aH3 = __builtin_amdgcn_wmma_f32_16x16x4_f32(false, aHi, false, b3, (short)0, aH3, false, false);
      }
      __syncthreads();                 // Bb[cur] consumed; Bb[cur^1] visible
    }

    // stage S tile: lane holds rows M = r + 8*hl, col N = ml (ISA C/D layout)
#pragma unroll
    for (int r = 0; r < 8; ++r) {
      int rl = wv * 32 + r + 8 * hl;
      sh.u.S[rl * SPITCH + 0 * 16 + ml] = aL0[r];
      sh.u.S[rl * SPITCH + 1 * 16 + ml] = aL1[r];
      sh.u.S[rl * SPITCH + 2 * 16 + ml] = aL2[r];
      sh.u.S[rl * SPITCH + 3 * 16 + ml] = aL3[r];
      int rh = rl + 16;
      sh.u.S[rh * SPITCH + 0 * 16 + ml] = aH0[r];
      sh.u.S[rh * SPITCH + 1 * 16 + ml] = aH1[r];
      sh.u.S[rh * SPITCH + 2 * 16 + ml] = aH2[r];
      sh.u.S[rh * SPITCH + 3 * 16 + ml] = aH3[r];
    }
    __syncthreads();

    // per-row distance + noisy top-5 (every thread owns one row)
    for (int c4 = 0; c4 < JBW; c4 += 4) {
      const v4f nz = __builtin_nontemporal_load((const v4f*)&nzrow[jb + c4]);
#pragma unroll
      for (int u = 0; u < 4; ++u) {
        const int j = jb + c4 + u;
        if (j == myrow) continue;                  // diagonal -> +inf
        float s  = sh.u.S[tid * SPITCH + c4 + u];
        float d2 = ni + xnorm2[j] - 2.f * s;       // xnorm2[j] wave-uniform
        float pd = sqrtf(fmaxf(d2, 1e-12f));
        float kn = pd + nz[u];                     // noisy ordering key
        if (kn < key[KNN - 1]) {
          key[KNN - 1] = kn; val[KNN - 1] = pd;    // clean value gathered
#pragma unroll
          for (int p = KNN - 1; p > 0; --p) {
            if (key[p] < key[p - 1]) {
              float tk = key[p]; key[p] = key[p - 1]; key[p - 1] = tk;
              float tv = val[p]; val[p] = val[p - 1]; val[p - 1] = tv;
            }
          }
        }
      }
    }
  }

#pragma unroll
  for (int i = 0; i < KNN; ++i) knn_out[(size_t)myrow * KNN + i] = val[i];
}

// ---------------------------------------------------------------------------
// Kernel 5: MLP fusion  feat[16] -> relu(@W1[16,64]+b1) -> @W2[64,32]+b2
// ---------------------------------------------------------------------------
__global__ void k_mlp(const float* __restrict__ assign,
                      const float* __restrict__ knn,
                      const float* __restrict__ stats,
                      const float* __restrict__ W1, const float* __restrict__ b1,
                      const float* __restrict__ W2, const float* __restrict__ b2,
                      float* __restrict__ enc) {
  __shared__ float sW1[16 * HDIM], sb1[HDIM], sW2[HDIM * ODIM], sb2[ODIM];
  const int tid = threadIdx.x;
  for (int i = tid; i < 16 * HDIM; i += blockDim.x) sW1[i] = W1[i];
  for (int i = tid; i < HDIM * ODIM; i += blockDim.x) sW2[i] = W2[i];
  if (tid < HDIM) sb1[tid] = b1[tid];
  if (tid < ODIM) sb2[tid] = b2[tid];
  __syncthreads();

  const int row = blockIdx.x * blockDim.x + tid;
  if (row >= BDIM) return;

  float f[16];
#pragma unroll
  for (int c = 0; c < CDIM; ++c) f[c] = assign[row * CDIM + c];
#pragma unroll
  for (int k = 0; k < KNN; ++k) f[CDIM + k] = knn[row * KNN + k];
#pragma unroll
  for (int s = 0; s < 3; ++s) f[CDIM + KNN + s] = stats[row * 3 + s];

  float h[HDIM];
#pragma unroll 8
  for (int j = 0; j < HDIM; ++j) {
    float a = sb1[j];
#pragma unroll
    for (int i = 0; i < 16; ++i) a += f[i] * sW1[i * HDIM + j];
    h[j] = fmaxf(a, 0.f);
  }
#pragma unroll 4
  for (int o = 0; o < ODIM; ++o) {
    float a = sb2[o];
#pragma unroll 16
    for (int j = 0; j < HDIM; ++j) a += h[j] * sW2[j * ODIM + o];
    enc[(size_t)row * ODIM + o] = a;
  }
}

// ---------------------------------------------------------------------------
// Kernel 6: deterministic loss finalize
// ---------------------------------------------------------------------------
__global__ void k_loss(const float* __restrict__ intra_part,
                       const float* __restrict__ inter_sum,
                       float* __restrict__ loss_out) {
  __shared__ float red[256];
  const int t = threadIdx.x;
  float s = 0.f;
  for (int i = t; i < BDIM; i += 256) s += intra_part[i];  // fixed order
  red[t] = s;
  __syncthreads();
  for (int off = 128; off > 0; off >>= 1) {
    if (t < off) red[t] += red[t + off];
    __syncthreads();
  }
  if (t == 0) {
    float intra = red[0] / (float)(BDIM * CDIM);
    float inter = inter_sum[0] / (float)(CDIM * (CDIM - 1));
    loss_out[0] = intra - 0.1f * inter;
  }
}

// ---------------------------------------------------------------------------
extern "C" void kernel_launch(void* const* d_in, const int* in_sizes, int n_in,
                              void* d_out, int out_size, void* d_ws, size_t ws_size,
                              hipStream_t stream) {
  const float* x     = (const float*)d_in[0];
  const float* cc    = (const float*)d_in[1];
  const float* temp  = (const float*)d_in[2];
  const float* cw    = (const float*)d_in[3];
  const float* W1    = (const float*)d_in[4];
  const float* b1    = (const float*)d_in[5];
  const float* W2    = (const float*)d_in[6];
  const float* b2    = (const float*)d_in[7];
  const float* noise = (const float*)d_in[8];

  float* out    = (float*)d_out;
  float* enc    = out;                           // [8192, 32]
  float* assign = enc + (size_t)BDIM * ODIM;     // [8192, 8]
  float* knn    = assign + (size_t)BDIM * CDIM;  // [8192, 5]
  float* stats  = knn + (size_t)BDIM * KNN;      // [8192, 3]
  float* loss   = stats + (size_t)BDIM * 3;      // [1]

  float* ws      = (float*)d_ws;
  float* xnorm2  = ws;            // [8192]
  float* intra   = ws + BDIM;     // [8192]
  float* cnorm   = ws + 2 * BDIM; // [8]
  float* inter   = cnorm + CDIM;  // [1]

  k_centers<<<1, 64, 0, stream>>>(cc, cnorm, inter);
  k_rowstats<<<BDIM / 8, 256, 0, stream>>>(x, stats, xnorm2);
  k_assign<<<BDIM / 8, 256, 0, stream>>>(x, cc, temp, cw, xnorm2, cnorm, assign, intra);
  k_gram_topk<<<BDIM / IB, 64, 0, stream>>>(x, noise, xnorm2, knn);
  k_mlp<<<BDIM / 256, 256, 0, stream>>>(assign, knn, stats, W1, b1, W2, b2, enc);
  k_loss<<<1, 256, 0, stream>>>(intra, inter, loss);
}